// AddDropMRR_21165598835552
// MI455X (gfx1250) — compile-verified
//
#include <hip/hip_runtime.h>
#include <math.h>
#include <stdint.h>

// ---------------------------------------------------------------------------
// Add-drop microring resonator, B x W elementwise with per-wavelength constants
//   through = A(l)*x + B(l)*y ; drop = C(l)*x + E*y   (complex coeffs, real x,y)
// outputs are |through| and |drop|, concatenated flat.
// ---------------------------------------------------------------------------

#define TPB 256
#define VEC 4
#define TILE_W (TPB * VEC)      // 1024 wavelengths per block tile
#define ROWS_PER_BLOCK 8        // rows streamed per block (coeff reuse)

// --- CDNA5 async global->LDS probing (compile-safe fallbacks) --------------
#if defined(__HIP_DEVICE_COMPILE__) && defined(__gfx1250__) && \
    __has_builtin(__builtin_amdgcn_global_load_async_to_lds_b128)
#define MRR_ASYNC_LDS 1
#else
#define MRR_ASYNC_LDS 0
#endif

#if MRR_ASYNC_LDS
// The builtin expects pointers to 4 x i32 vectors (per hipcc diagnostic):
//   param0: v4i in global/device address space (1)
//   param1: v4i in LDS address space (3)
typedef int mrr_v4i __attribute__((ext_vector_type(4)));
typedef __attribute__((address_space(1))) mrr_v4i* mrr_gv4i_p;
typedef __attribute__((address_space(3))) mrr_v4i* mrr_lv4i_p;

#if __has_builtin(__builtin_amdgcn_s_wait_asynccnt)
#define MRR_WAIT_ASYNC() __builtin_amdgcn_s_wait_asynccnt(0)
#else
#define MRR_WAIT_ASYNC() asm volatile("s_wait_asynccnt 0x0" ::: "memory")
#endif
#endif

// ---------------------------------------------------------------------------
// Kernel 1: per-wavelength complex coefficients (W elements, trivial cost)
// ---------------------------------------------------------------------------
__global__ void AddDropMRR_constants(
    const float* __restrict__ wavelengths,
    const float* __restrict__ coupling_1,
    const float* __restrict__ coupling_2,
    const float* __restrict__ phi_1,
    const float* __restrict__ phi_2,
    const float* __restrict__ phi_ring,
    const float* __restrict__ alpha,
    float4* __restrict__ cAB,   // {A_r, A_i, B_r, B_i}
    float4* __restrict__ cCE,   // {C_r, C_i, E_r, E_i}
    int W)
{
    int w = blockIdx.x * blockDim.x + threadIdx.x;
    if (w >= W) return;

    const float k1 = coupling_1[0];
    const float k2 = coupling_2[0];
    const float p1 = phi_1[0];
    const float p2 = phi_2[0];
    const float pr = phi_ring[0];
    const float al = alpha[0];

    const float k1c = fminf(fmaxf(k1, 0.01f), 0.99f);
    const float k2c = fminf(fmaxf(k2, 0.01f), 0.99f);
    const float t1  = sqrtf(fmaxf(1.0f - k1c * k1c, 0.0f));
    const float t2  = sqrtf(fmaxf(1.0f - k2c * k2c, 0.0f));
    const float sq1 = sqrtf(k1);   // raw (unclamped), as in reference
    const float sq2 = sqrtf(k2);

    float s1e, c1e; __sincosf(p1, &s1e, &c1e);
    float s2e, c2e; __sincosf(p2, &s2e, &c2e);
    // g = j*k1c*e^{j p1},  D = j*k2c*e^{j p2}
    const float g_r = -k1c * s1e, g_i = k1c * c1e;
    const float d_r = -k2c * s2e, d_i = k2c * c2e;
    // resonance denominator constant (raw couplings, as in reference)
    const float denc = al * sqrtf(1.0f - k1 * k1) * sqrtf(1.0f - k2 * k2);

    const float TWO_PI_NEFF = 15.07964474f;     // 2*pi*2.4
    const float CIRC        = 6.283185307e-5f;  // 2*pi*1e-5
    const float lam = wavelengths[w];
    const float phi_rt = (TWO_PI_NEFF / lam) * CIRC + pr;
    float sph, cph; __sincosf(phi_rt, &sph, &cph);

    // R = alpha * g * exp_phi
    const float Rr = al * (g_r * cph - g_i * sph);
    const float Ri = al * (g_r * sph + g_i * cph);
    // den = 1 - denc * exp_phi
    const float den_r = 1.0f - denc * cph;
    const float den_i = -denc * sph;
    // S = t2*sqrt(c1)/den  (complex reciprocal)
    const float inv = (t2 * sq1) / fmaf(den_r, den_r, den_i * den_i);
    const float S_r = inv * den_r;
    const float S_i = -inv * den_i;
    // A = t1 + S*R ; B = S*sqrt(c2)
    const float A_r = t1 + S_r * Rr - S_i * Ri;
    const float A_i = S_r * Ri + S_i * Rr;
    const float B_r = S_r * sq2;
    const float B_i = S_i * sq2;
    // C = D*R ; E = D*sqrt(c2)
    const float C_r = d_r * Rr - d_i * Ri;
    const float C_i = d_r * Ri + d_i * Rr;
    const float E_r = d_r * sq2;
    const float E_i = d_i * sq2;

    cAB[w] = make_float4(A_r, A_i, B_r, B_i);
    cCE[w] = make_float4(C_r, C_i, E_r, E_i);
}

// ---------------------------------------------------------------------------
// Kernel 2: streaming elementwise evaluation (HBM bound)
// ---------------------------------------------------------------------------
__device__ __forceinline__ void mrr_eval(float xs, float ys,
                                         float4 ab, float4 ce,
                                         float& to, float& dr)
{
    float tr = fmaf(ab.x, xs, ab.z * ys);
    float ti = fmaf(ab.y, xs, ab.w * ys);
    to = sqrtf(fmaf(tr, tr, ti * ti));
    float er = fmaf(ce.x, xs, ce.z * ys);
    float ei = fmaf(ce.y, xs, ce.w * ys);
    dr = sqrtf(fmaf(er, er, ei * ei));
}

__global__ __launch_bounds__(TPB) void AddDropMRR_main(
    const float* __restrict__ xg,    // input_signal [B,W]
    const float* __restrict__ yg,    // add_signal   [B,W]
    const float4* __restrict__ cAB,
    const float4* __restrict__ cCE,
    float* __restrict__ out_thr,
    float* __restrict__ out_drp,
    int W, int B)
{
    __shared__ float4 sAB[TILE_W];
    __shared__ float4 sCE[TILE_W];

    const int lane4 = threadIdx.x * VEC;          // lambda offset in tile
    const int col   = blockIdx.x * TILE_W + lane4;
    if (col >= W) return;                          // W % TILE_W == 0 in practice

    // --- stage this thread's 4 lambdas of coefficients into LDS ------------
#if MRR_ASYNC_LDS
    {
        const float4* gAB = cAB + col;
        const float4* gCE = cCE + col;
        float4* lAB = &sAB[lane4];
        float4* lCE = &sCE[lane4];
#pragma unroll
        for (int i = 0; i < VEC; ++i) {
            __builtin_amdgcn_global_load_async_to_lds_b128(
                (mrr_gv4i_p)(gAB + i), (mrr_lv4i_p)(lAB + i), 0, 0);
            __builtin_amdgcn_global_load_async_to_lds_b128(
                (mrr_gv4i_p)(gCE + i), (mrr_lv4i_p)(lCE + i), 0, 0);
        }
        MRR_WAIT_ASYNC();   // each thread only reads LDS it wrote itself
    }
#else
#pragma unroll
    for (int i = 0; i < VEC; ++i) {
        sAB[lane4 + i] = cAB[col + i];
        sCE[lane4 + i] = cCE[col + i];
    }
#endif

    // pull coefficients into registers once; reuse across all rows
    float4 ab[VEC], ce[VEC];
#pragma unroll
    for (int i = 0; i < VEC; ++i) {
        ab[i] = sAB[lane4 + i];
        ce[i] = sCE[lane4 + i];
    }

    const int w4   = W >> 2;
    const int c4   = col >> 2;
    const int row0 = blockIdx.y * ROWS_PER_BLOCK;
    const int rend = min(row0 + ROWS_PER_BLOCK, B);

    const float4* x4 = (const float4*)xg;
    const float4* y4 = (const float4*)yg;
    float4* t4 = (float4*)out_thr;
    float4* d4 = (float4*)out_drp;

    for (int row = row0; row < rend; ++row) {
        size_t idx = (size_t)row * (size_t)w4 + (size_t)c4;
        if (row + 1 < rend) {   // hint next row into cache (global_prefetch)
            __builtin_prefetch(&x4[idx + w4], 0, 0);
            __builtin_prefetch(&y4[idx + w4], 0, 0);
        }
        float4 xv = x4[idx];
        float4 yv = y4[idx];
        float4 to, dr;
        mrr_eval(xv.x, yv.x, ab[0], ce[0], to.x, dr.x);
        mrr_eval(xv.y, yv.y, ab[1], ce[1], to.y, dr.y);
        mrr_eval(xv.z, yv.z, ab[2], ce[2], to.z, dr.z);
        mrr_eval(xv.w, yv.w, ab[3], ce[3], to.w, dr.w);
        t4[idx] = to;
        d4[idx] = dr;
    }
}

// ---------------------------------------------------------------------------
extern "C" void kernel_launch(void* const* d_in, const int* in_sizes, int n_in,
                              void* d_out, int out_size, void* d_ws, size_t ws_size,
                              hipStream_t stream)
{
    const float* input_signal = (const float*)d_in[0];
    const float* add_signal   = (const float*)d_in[1];
    const float* wavelengths  = (const float*)d_in[2];
    const float* coupling_1   = (const float*)d_in[3];
    const float* coupling_2   = (const float*)d_in[4];
    const float* phi_1        = (const float*)d_in[5];
    const float* phi_2        = (const float*)d_in[6];
    const float* phi_ring     = (const float*)d_in[7];
    const float* alpha        = (const float*)d_in[8];

    const int W = in_sizes[2];
    const int B = in_sizes[0] / W;

    float4* cAB = (float4*)d_ws;        // W float4s
    float4* cCE = cAB + W;              // W float4s  (total 32*W bytes)

    {
        dim3 grid((W + TPB - 1) / TPB);
        AddDropMRR_constants<<<grid, TPB, 0, stream>>>(
            wavelengths, coupling_1, coupling_2, phi_1, phi_2, phi_ring, alpha,
            cAB, cCE, W);
    }
    {
        dim3 grid((W + TILE_W - 1) / TILE_W,
                  (B + ROWS_PER_BLOCK - 1) / ROWS_PER_BLOCK);
        float* out_thr = (float*)d_out;
        float* out_drp = out_thr + (size_t)B * (size_t)W;
        AddDropMRR_main<<<grid, TPB, 0, stream>>>(
            input_signal, add_signal, cAB, cCE, out_thr, out_drp, W, B);
    }
}